// BFPQuantizerSTE_919123001649
// MI455X (gfx1250) — compile-verified
//
#include <hip/hip_runtime.h>

// Block floating-point quantize/dequantize, BLOCK_SIZE=16, MANTISSA_BITS=8.
// Memory-bound streaming kernel: 1 GiB traffic -> ~46us at 23.3 TB/s HBM.
// Layout: 1 lane = 1 float4 (b128), 4 consecutive lanes = one 16-elem block.

typedef float f32x4 __attribute__((ext_vector_type(4)));

// Quad-wide max via ds_swizzle_b32 immediates (group-of-32 mode:
// offset = xor_mask<<10 | or_mask<<5 | and_mask). No index VGPRs needed.
#define SWZ_SWAPX1 0x041F  // xor=1, and=0x1f
#define SWZ_SWAPX2 0x081F  // xor=2, and=0x1f

__device__ __forceinline__ float quad_max(float a) {
    float t = __int_as_float(__builtin_amdgcn_ds_swizzle(__float_as_int(a), SWZ_SWAPX1));
    a = fmaxf(a, t);
    t = __int_as_float(__builtin_amdgcn_ds_swizzle(__float_as_int(a), SWZ_SWAPX2));
    return fmaxf(a, t);
}

__device__ __forceinline__ float qdq_one(float x, int shift) {
    // exact scale by 2^shift, round-half-even, clamp to [-128,127], scale back
    float s = ldexpf(x, shift);          // v_ldexp_f32 (exact)
    float q = rintf(s);                  // v_rndne_f32 (round half to even)
    q = fminf(fmaxf(q, -128.0f), 127.0f);
    return ldexpf(q, -shift);            // exact dequantize
}

__global__ __launch_bounds__(256) void bfp_qdq_kernel(const float* __restrict__ in,
                                                      float* __restrict__ out,
                                                      long long n4) {
    const long long stride = (long long)gridDim.x * (long long)blockDim.x;
    long long i = (long long)blockIdx.x * blockDim.x + threadIdx.x;
    for (; i < n4; i += stride) {
        // Speculative prefetch one grid-stride ahead (global_prefetch_b8;
        // OOB-safe: translation failures are silently dropped).
        __builtin_prefetch((const char*)(in + 4 * (i + stride)), 0, 0);

        // Streaming load: data is touched exactly once and >> L2, so NT hint.
        f32x4 v = __builtin_nontemporal_load((const f32x4*)in + i);

        // Lane-local max|.| over 4 elements, then block max across the quad
        // of lanes sharing this 16-elem block. Quads are always uniformly
        // active: n4, the grid stride, and quad base indices are all
        // multiples of 4, so EXEC divergence never splits a quad.
        float a = fmaxf(fmaxf(fabsf(v.x), fabsf(v.y)),
                        fmaxf(fabsf(v.z), fabsf(v.w)));
        a = quad_max(a);

        // Shared exponent: a = f * 2^ex, f in [0.5,1) => floor(log2 a) = ex-1.
        // Denormal-safe via v_frexp_exp_i32_f32. a==0 -> all-zero block flows
        // through to q=0 -> out=0, matching the reference's zero-block path.
        int ex;
        (void)frexpf(a, &ex);
        const int shift = 8 - ex;        // 7 - e, with e = ex - 1

        f32x4 r;
        r.x = qdq_one(v.x, shift);
        r.y = qdq_one(v.y, shift);
        r.z = qdq_one(v.z, shift);
        r.w = qdq_one(v.w, shift);

        // Streaming store with NT hint.
        __builtin_nontemporal_store(r, (f32x4*)out + i);
    }
}

extern "C" void kernel_launch(void* const* d_in, const int* in_sizes, int n_in,
                              void* d_out, int out_size, void* d_ws, size_t ws_size,
                              hipStream_t stream) {
    (void)n_in; (void)d_ws; (void)ws_size; (void)out_size;
    const float* x = (const float*)d_in[0];
    float* out = (float*)d_out;

    const long long n  = (long long)in_sizes[0];   // 8*4096*4096 = 134,217,728
    const long long n4 = n / 4;                    // one float4 per lane-iter

    // ~8 float4 iterations per thread; tens of thousands of wave32 waves in
    // flight to saturate HBM while keeping the grid-stride prefetch useful.
    const int threads = 256;
    long long want_blocks = (n4 + (long long)threads * 8 - 1) / ((long long)threads * 8);
    if (want_blocks < 1) want_blocks = 1;
    if (want_blocks > 1048576) want_blocks = 1048576;
    const int blocks = (int)want_blocks;

    bfp_qdq_kernel<<<blocks, threads, 0, stream>>>(x, out, n4);
}